// DotProductAttention_16183436771978
// MI455X (gfx1250) — compile-verified
//
#include <hip/hip_runtime.h>
#include <hip/hip_bf16.h>
#include <math.h>

#define NN 50000
#define NE 800000
#define EDGE_TPB 8   // edge tiles (of 16 edges) per block; 50000/8 = 6250 exact
#define NODE_TPB 8   // node tiles per block (with uniform bounds check)

typedef __attribute__((ext_vector_type(2))) float v2f;
typedef __attribute__((ext_vector_type(8))) float v8f;

// ---------- WMMA helpers (fp32, 16x16x4) ----------
__device__ __forceinline__ v8f wmma_k4(v2f a, v2f b, v8f c) {
  return __builtin_amdgcn_wmma_f32_16x16x4_f32(false, a, false, b, (short)0, c, false, false);
}

// Preload one wave's B fragments (16 k-steps) for a 16-col slice of W (row-major K x ldb)
__device__ __forceinline__ void load_bfrag(v2f* B, const float* __restrict__ W,
                                           int ldb, int colBase, int lane) {
  const int half = lane >> 4;
  const int col = colBase + (lane & 15);
#pragma unroll
  for (int kk = 0; kk < 16; ++kk) {
    const int kb = kk * 4 + half * 2;
    B[kk].x = W[(size_t)kb * ldb + col];
    B[kk].y = W[(size_t)(kb + 1) * ldb + col];
  }
}

// D(16x16) = A_lds(16x64 row-major) x B(register fragments) + 0
__device__ __forceinline__ v8f gemm_tile_reg(const float* Alds, const v2f* B, int lane) {
  v8f acc = {};
  const int row = lane & 15;
  const int half = lane >> 4;
#pragma unroll
  for (int kk = 0; kk < 16; ++kk) {
    v2f a = *(const v2f*)(Alds + row * 64 + kk * 4 + half * 2);
    acc = wmma_k4(a, B[kk], acc);
  }
  return acc;
}

// ---------- order-preserving float<->uint for atomicMax ----------
__device__ __forceinline__ unsigned int fmax_enc(float f) {
  unsigned int u = __float_as_uint(f);
  return (u & 0x80000000u) ? ~u : (u | 0x80000000u);
}
__device__ __forceinline__ float fmax_dec(unsigned int e) {
  unsigned int u = (e & 0x80000000u) ? (e ^ 0x80000000u) : ~e;
  return __uint_as_float(u);
}

__device__ __forceinline__ float silu(float x) { return x / (1.0f + __expf(-x)); }

// ---------- init workspace (graph-replay safe: runs every launch) ----------
__global__ void init_ws_kernel(unsigned int* segmax, float* denom, float* attn) {
  int i = blockIdx.x * blockDim.x + threadIdx.x;
  if (i < NN * 4) { segmax[i] = 0x007FFFFFu; /* enc(-inf) */ denom[i] = 0.0f; }
  if (i < NN * 64) attn[i] = 0.0f;
}

// ---------- node-side linears: q (scaled), msg_src, msg_dst ----------
__global__ void __launch_bounds__(128)
node_linear_kernel(const float* __restrict__ node_input,
                   const float* __restrict__ Wq,    const float* __restrict__ bq,
                   const float* __restrict__ Wsrc,  const float* __restrict__ bsrc,
                   const float* __restrict__ Wdst,
                   float* __restrict__ q_ws, float* __restrict__ ms_ws, float* __restrict__ md_ws) {
  __shared__ float A[16 * 64];
  const int t = threadIdx.x;
  const int lane = t & 31, w = t >> 5, half = lane >> 4, ln = lane & 15;
  const int col = w * 16 + ln;

  v2f Bq[16], Bs[16], Bd[16];
  load_bfrag(Bq, Wq,   64, w * 16, lane);
  load_bfrag(Bs, Wsrc, 64, w * 16, lane);
  load_bfrag(Bd, Wdst, 64, w * 16, lane);
  const float biasq = bq[col];
  const float biass = bsrc[col];

  const int nodeTiles = NN / 16;
  for (int it = 0; it < NODE_TPB; ++it) {
    const int tile = blockIdx.x * NODE_TPB + it;
    if (tile >= nodeTiles) break;          // block-uniform
    const int base = tile * 16;
    __syncthreads();                       // guard A reuse across iterations
    {
      const int row = t >> 3, c0 = (t & 7) * 8;
      const float4* p = (const float4*)(node_input + (size_t)(base + row) * 64 + c0);
      *(float4*)(A + row * 64 + c0)     = p[0];
      *(float4*)(A + row * 64 + c0 + 4) = p[1];
    }
    __syncthreads();
    {
      v8f d = gemm_tile_reg(A, Bq, lane);
#pragma unroll
      for (int r = 0; r < 8; ++r)
        q_ws[(size_t)(base + r + 8 * half) * 64 + col] = (d[r] + biasq) * 0.25f; // 1/sqrt(16)
    }
    {
      v8f d = gemm_tile_reg(A, Bs, lane);
#pragma unroll
      for (int r = 0; r < 8; ++r)
        ms_ws[(size_t)(base + r + 8 * half) * 64 + col] = d[r] + biass;
    }
    {
      v8f d = gemm_tile_reg(A, Bd, lane);
#pragma unroll
      for (int r = 0; r < 8; ++r)
        md_ws[(size_t)(base + r + 8 * half) * 64 + col] = d[r];
    }
  }
}

// ---------- edge pipeline: MLP -> tp -> kv2 half; PASS 1 = k/alpha/max, PASS 2 = v/scatter ----------
template <int PASS>
__global__ void __launch_bounds__(128)
edge_kernel(const int* __restrict__ edge_src, const int* __restrict__ edge_dst,
            const float* __restrict__ edge_attr, const float* __restrict__ edge_scalars,
            const float* __restrict__ Wfc1, const float* __restrict__ bfc1,
            const float* __restrict__ Wfc2, const float* __restrict__ bfc2,
            const float* __restrict__ Wfc3, const float* __restrict__ bfc3,
            const float* __restrict__ Wkv,  const float* __restrict__ bkv,
            const float* __restrict__ ms_ws, const float* __restrict__ md_ws,
            const float* __restrict__ q_ws,
            float* __restrict__ alpha_ws, unsigned int* __restrict__ segmax_ws,
            float* __restrict__ denom_ws, float* __restrict__ attn_ws) {
  __shared__ float bufA[16 * 64];
  __shared__ float bufB[16 * 64];
  __shared__ float kvb[16 * 64];
  __shared__ int   src_s[16];
  __shared__ int   dst_s[16];
  __shared__ float ea_s[16];
  __shared__ float ex_s[4 * 16];

  const int t = threadIdx.x;
  const int lane = t & 31, w = t >> 5, half = lane >> 4, ln = lane & 15;

  // Weights live in registers for the whole block (128 VGPRs of B fragments)
  v2f B1[16], B2[16], B3[16], B4[16];
  load_bfrag(B1, Wfc1, 64, w * 16, lane);
  load_bfrag(B2, Wfc2, 64, w * 16, lane);
  load_bfrag(B3, Wfc3, 64, w * 16, lane);
  const int kvCol = (PASS == 1) ? (w * 16) : (64 + w * 16);
  load_bfrag(B4, Wkv, 128, kvCol, lane);
  const float bias1 = bfc1[w * 16 + ln];
  const float bias2 = bfc2[w * 16 + ln];
  const float bias3 = bfc3[w * 16 + ln];
  const float bias4 = bkv[kvCol + ln];

  for (int it = 0; it < EDGE_TPB; ++it) {
    const int ebase = (blockIdx.x * EDGE_TPB + it) * 16;
    __syncthreads();                       // guard shared reuse across iterations
    if (t < 16) {
      src_s[t] = edge_src[ebase + t];
      dst_s[t] = edge_dst[ebase + t];
      ea_s[t]  = edge_attr[ebase + t];
    }
    __syncthreads();

    { // gather kv tile = msg_src[src] + msg_dst[dst]; stage edge_scalars tile
      const int row = t >> 3, c0 = (t & 7) * 8;
      const float4* ps = (const float4*)(ms_ws + (size_t)src_s[row] * 64 + c0);
      const float4* pd = (const float4*)(md_ws + (size_t)dst_s[row] * 64 + c0);
      float4 a0 = ps[0], a1 = ps[1], b0 = pd[0], b1 = pd[1];
      float4 r0, r1;
      r0.x = a0.x + b0.x; r0.y = a0.y + b0.y; r0.z = a0.z + b0.z; r0.w = a0.w + b0.w;
      r1.x = a1.x + b1.x; r1.y = a1.y + b1.y; r1.z = a1.z + b1.z; r1.w = a1.w + b1.w;
      *(float4*)(kvb + row * 64 + c0)     = r0;
      *(float4*)(kvb + row * 64 + c0 + 4) = r1;
      const float4* pe = (const float4*)(edge_scalars + (size_t)(ebase + row) * 64 + c0);
      *(float4*)(bufA + row * 64 + c0)     = pe[0];
      *(float4*)(bufA + row * 64 + c0 + 4) = pe[1];
    }
    __syncthreads();

    { // h1 = silu(es @ Wfc1 + b)
      v8f d = gemm_tile_reg(bufA, B1, lane);
#pragma unroll
      for (int r = 0; r < 8; ++r)
        bufB[(r + 8 * half) * 64 + w * 16 + ln] = silu(d[r] + bias1);
    }
    __syncthreads();
    { // h2 = silu(h1 @ Wfc2 + b)
      v8f d = gemm_tile_reg(bufB, B2, lane);
#pragma unroll
      for (int r = 0; r < 8; ++r)
        bufA[(r + 8 * half) * 64 + w * 16 + ln] = silu(d[r] + bias2);
    }
    __syncthreads();
    { // tp = kv * edge_attr * (h2 @ Wfc3 + b)
      v8f d = gemm_tile_reg(bufA, B3, lane);
#pragma unroll
      for (int r = 0; r < 8; ++r) {
        const int row = r + 8 * half;
        bufB[row * 64 + w * 16 + ln] =
            kvb[row * 64 + w * 16 + ln] * ea_s[row] * (d[r] + bias3);
      }
    }
    __syncthreads();

    if (PASS == 1) {
      // k head w = (tp @ Wkv)[:, w*16:+16] ; alpha = <q_scaled[dst], k>
      v8f dk = gemm_tile_reg(bufB, B4, lane);
#pragma unroll
      for (int r = 0; r < 8; ++r) {
        const int row = r + 8 * half;
        const float qv = q_ws[(size_t)dst_s[row] * 64 + w * 16 + ln];
        float p = (dk[r] + bias4) * qv;
        p += __shfl_xor(p, 1, 16);
        p += __shfl_xor(p, 2, 16);
        p += __shfl_xor(p, 4, 16);
        p += __shfl_xor(p, 8, 16);
        if (ln == 0) {
          alpha_ws[(size_t)(ebase + row) * 4 + w] = p;
          atomicMax(&segmax_ws[(size_t)dst_s[row] * 4 + w], fmax_enc(p));
        }
      }
    } else {
      // v head w = (tp @ Wkv)[:, 64+w*16:+16]; scatter v*exp(alpha-max)
      v8f dv = gemm_tile_reg(bufB, B4, lane);
      if (lane < 16) {
        const int row = lane;
        float m = fmax_dec(segmax_ws[(size_t)dst_s[row] * 4 + w]);
        if (!isfinite(m)) m = 0.0f;
        const float ex = __expf(alpha_ws[(size_t)(ebase + row) * 4 + w] - m);
        ex_s[w * 16 + row] = ex;
        atomicAdd(&denom_ws[(size_t)dst_s[row] * 4 + w], ex);
      }
      __syncthreads();
#pragma unroll
      for (int r = 0; r < 8; ++r) {
        const int row = r + 8 * half;
        const float vv = (dv[r] + bias4) * ex_s[w * 16 + row];
        atomicAdd(&attn_ws[(size_t)dst_s[row] * 64 + w * 16 + ln], vv);
      }
    }
  }
}

// ---------- finalize: out = (attn / denom) @ W_proj + b_proj ----------
__global__ void __launch_bounds__(128)
out_kernel(const float* __restrict__ attn_ws, const float* __restrict__ denom_ws,
           const float* __restrict__ Wproj, const float* __restrict__ bproj,
           float* __restrict__ out) {
  __shared__ float A[16 * 64];
  const int t = threadIdx.x;
  const int lane = t & 31, w = t >> 5, half = lane >> 4, ln = lane & 15;

  v2f Bp[16];
  load_bfrag(Bp, Wproj, 64, w * 16, lane);
  const float bias = bproj[w * 16 + ln];

  const int nodeTiles = NN / 16;
  for (int it = 0; it < NODE_TPB; ++it) {
    const int tile = blockIdx.x * NODE_TPB + it;
    if (tile >= nodeTiles) break;          // block-uniform
    const int base = tile * 16;
    __syncthreads();
    {
      const int row = t >> 3, c0 = (t & 7) * 8;
#pragma unroll
      for (int i = 0; i < 8; ++i) {
        const int c = c0 + i;
        const float dnm = denom_ws[(size_t)(base + row) * 4 + (c >> 4)] + 1e-16f;
        A[row * 64 + c] = attn_ws[(size_t)(base + row) * 64 + c] / dnm;
      }
    }
    __syncthreads();
    v8f d = gemm_tile_reg(A, Bp, lane);
#pragma unroll
    for (int r = 0; r < 8; ++r)
      out[(size_t)(base + r + 8 * half) * 64 + w * 16 + ln] = d[r] + bias;
  }
}

extern "C" void kernel_launch(void* const* d_in, const int* in_sizes, int n_in,
                              void* d_out, int out_size, void* d_ws, size_t ws_size,
                              hipStream_t stream) {
  const float* node_input   = (const float*)d_in[0];
  // d_in[1] node_attr (unused)
  const int*   edge_src     = (const int*)d_in[2];
  const int*   edge_dst     = (const int*)d_in[3];
  const float* edge_attr    = (const float*)d_in[4];
  const float* edge_scalars = (const float*)d_in[5];
  // d_in[6] batch (unused)
  const float* Wq    = (const float*)d_in[7];
  const float* bq    = (const float*)d_in[8];
  const float* Wsrc  = (const float*)d_in[9];
  const float* bsrc  = (const float*)d_in[10];
  const float* Wdst  = (const float*)d_in[11];
  const float* Wfc1  = (const float*)d_in[12];
  const float* bfc1  = (const float*)d_in[13];
  const float* Wfc2  = (const float*)d_in[14];
  const float* bfc2  = (const float*)d_in[15];
  const float* Wfc3  = (const float*)d_in[16];
  const float* bfc3  = (const float*)d_in[17];
  const float* Wkv   = (const float*)d_in[18];
  const float* bkv   = (const float*)d_in[19];
  const float* Wproj = (const float*)d_in[20];
  const float* bproj = (const float*)d_in[21];
  float* out = (float*)d_out;

  // Workspace carve-up (~66 MB total)
  float* q_ws     = (float*)d_ws;                 // NN*64
  float* ms_ws    = q_ws  + (size_t)NN * 64;      // NN*64
  float* md_ws    = ms_ws + (size_t)NN * 64;      // NN*64
  float* alpha_ws = md_ws + (size_t)NN * 64;      // NE*4
  float* denom_ws = alpha_ws + (size_t)NE * 4;    // NN*4
  float* attn_ws  = denom_ws + (size_t)NN * 4;    // NN*64
  unsigned int* segmax_ws = (unsigned int*)(attn_ws + (size_t)NN * 64); // NN*4

  const int nodeBlocks = (NN / 16 + NODE_TPB - 1) / NODE_TPB;  // 391
  const int edgeBlocks = NE / 16 / EDGE_TPB;                   // 6250

  init_ws_kernel<<<(NN * 64 + 255) / 256, 256, 0, stream>>>(segmax_ws, denom_ws, attn_ws);

  node_linear_kernel<<<nodeBlocks, 128, 0, stream>>>(
      node_input, Wq, bq, Wsrc, bsrc, Wdst, q_ws, ms_ws, md_ws);

  edge_kernel<1><<<edgeBlocks, 128, 0, stream>>>(
      edge_src, edge_dst, edge_attr, edge_scalars,
      Wfc1, bfc1, Wfc2, bfc2, Wfc3, bfc3, Wkv, bkv,
      ms_ws, md_ws, q_ws, alpha_ws, segmax_ws, denom_ws, attn_ws);

  edge_kernel<2><<<edgeBlocks, 128, 0, stream>>>(
      edge_src, edge_dst, edge_attr, edge_scalars,
      Wfc1, bfc1, Wfc2, bfc2, Wfc3, bfc3, Wkv, bkv,
      ms_ws, md_ws, q_ws, alpha_ws, segmax_ws, denom_ws, attn_ws);

  out_kernel<<<nodeBlocks, 128, 0, stream>>>(attn_ws, denom_ws, Wproj, bproj, out);
}